// KGPolicy_84894323573127
// MI455X (gfx1250) — compile-verified
//
#include <hip/hip_runtime.h>

typedef float v2f __attribute__((ext_vector_type(2)));
typedef float v8f __attribute__((ext_vector_type(8)));

constexpr int kUsers = 150000;
constexpr int kEnt   = 250000;
constexpr int kCh    = 64;
constexpr int kRel   = 16;     // N_REL - 1
constexpr int kF     = 4;
constexpr int kEdges = 1000000;
constexpr int kNnz   = 1000000;

// ---------------------------------------------------------------------------
// Tiny precompute: cor scalar + disen_weight = softmax(att) @ weight  [4][64]
// ---------------------------------------------------------------------------
__global__ void k_small(const float* __restrict__ att,      // [4][16]
                        const float* __restrict__ weight,   // [16][64]
                        float* __restrict__ disen_w,        // [4][64]
                        float* __restrict__ cor_out) {
  int c = threadIdx.x;                     // 0..63
  float p[kF][kRel];
  for (int f = 0; f < kF; ++f) {
    float m = -1e30f;
    for (int r = 0; r < kRel; ++r) m = fmaxf(m, att[f * kRel + r]);
    float s = 0.f;
    for (int r = 0; r < kRel; ++r) { float e = expf(att[f * kRel + r] - m); p[f][r] = e; s += e; }
    float inv = 1.f / s;
    for (int r = 0; r < kRel; ++r) p[f][r] *= inv;
  }
  if (c < kCh) {
    for (int f = 0; f < kF; ++f) {
      float acc = 0.f;
      for (int r = 0; r < kRel; ++r) acc += p[f][r] * weight[r * kCh + c];
      disen_w[f * kCh + c] = acc;
    }
  }
  if (c == 0) {
    float nrm[kF];
    for (int f = 0; f < kF; ++f) {
      float s = 0.f;
      for (int r = 0; r < kRel; ++r) s += att[f * kRel + r] * att[f * kRel + r];
      nrm[f] = sqrtf(s);
    }
    float cor = 0.f;
    for (int i = 0; i < kF; ++i)
      for (int j = i + 1; j < kF; ++j) {
        float d = 0.f;
        for (int r = 0; r < kRel; ++r) d += att[i * kRel + r] * att[j * kRel + r];
        float ct = d / (nrm[i] * nrm[j]);
        cor += ct * ct;
      }
    *cor_out = cor;
  }
}

// ---------------------------------------------------------------------------
// Degree counts of KG heads (f32 so denom = max(count,1) is trivial later)
// ---------------------------------------------------------------------------
__global__ void k_count(const int* __restrict__ head, float* __restrict__ counts) {
  int i = blockIdx.x * blockDim.x + threadIdx.x;
  if (i < kEdges) atomicAdd(&counts[head[i]], 1.0f);
}

// ---------------------------------------------------------------------------
// KG edge scatter: e_agg[head] += e_prev[tail] * weight[rel]
// 16 lanes per edge, float4 per lane, f32 atomics (hit mostly in L2: the
// 64MB entity table fits in the 192MB global L2).
// ---------------------------------------------------------------------------
__global__ void k_kg_scatter(const float* __restrict__ e_prev,
                             const float* __restrict__ weight,
                             const int* __restrict__ edge_index,
                             const int* __restrict__ edge_type,
                             float* __restrict__ e_agg) {
  int idx = blockIdx.x * blockDim.x + threadIdx.x;
  int edge = idx >> 4;
  if (edge >= kEdges) return;
  int q = (idx & 15) << 2;
  int h = edge_index[edge];
  int t = edge_index[kEdges + edge];
  int r = edge_type[edge] - 1;
  float4 ev = *reinterpret_cast<const float4*>(&e_prev[(size_t)t * kCh + q]);
  float4 w  = *reinterpret_cast<const float4*>(&weight[(size_t)r * kCh + q]);
  float* dst = &e_agg[(size_t)h * kCh + q];
  atomicAdd(dst + 0, ev.x * w.x);
  atomicAdd(dst + 1, ev.y * w.y);
  atomicAdd(dst + 2, ev.z * w.z);
  atomicAdd(dst + 3, ev.w * w.w);
}

// ---------------------------------------------------------------------------
// User interaction scatter: u_agg[row] += val * e_prev[col]
// ---------------------------------------------------------------------------
__global__ void k_inter_scatter(const float* __restrict__ e_prev,
                                const float* __restrict__ val,
                                const int* __restrict__ row,
                                const int* __restrict__ col,
                                float* __restrict__ u_agg) {
  int idx = blockIdx.x * blockDim.x + threadIdx.x;
  int i = idx >> 4;
  if (i >= kNnz) return;
  int q = (idx & 15) << 2;
  int rr = row[i];
  int cc = col[i];
  float v = val[i];
  float4 ev = *reinterpret_cast<const float4*>(&e_prev[(size_t)cc * kCh + q]);
  float* dst = &u_agg[(size_t)rr * kCh + q];
  atomicAdd(dst + 0, v * ev.x);
  atomicAdd(dst + 1, v * ev.y);
  atomicAdd(dst + 2, v * ev.z);
  atomicAdd(dst + 3, v * ev.w);
}

// ---------------------------------------------------------------------------
// User finalize (WMMA): one wave handles 16 users.
//   scores  = softmax(u_prev @ latent^T)              (per-lane FMAs, K=64,F=4)
//   mix     = scores @ disen_weight                    V_WMMA_F32_16X16X4_F32
//   u_new   = l2norm(u_agg * (1 + mix))               (in place into u_agg)
//   usr_res += u_new
// A 16x4 f32 layout: lanes 0-15 {K0,K1}, lanes 16-31 {K2,K3} (M = lane&15).
// B 4x16 f32 layout: mirrored (lanes 0-15 rows {0,1}, lanes 16-31 rows {2,3}).
// D 16x16 f32: vgpr j, lanes 0-15 -> (M=j, N=lane), lanes 16-31 -> (M=8+j).
// All control flow before WMMA is wave-uniform => EXEC all ones.
// ---------------------------------------------------------------------------
__global__ void __launch_bounds__(256)
k_user_finalize(const float* __restrict__ u_prev,
                float* __restrict__ u_agg,
                const float* __restrict__ latent,   // [4][64]
                const float* __restrict__ disen_w,  // [4][64]
                float* __restrict__ usr_res) {
  __shared__ float s_lat[kF * kCh];
  __shared__ float s_dw[kF * kCh];
  s_lat[threadIdx.x] = latent[threadIdx.x];   // blockDim.x == 256 == kF*kCh
  s_dw[threadIdx.x]  = disen_w[threadIdx.x];
  __syncthreads();

  int gwave = (blockIdx.x * blockDim.x + threadIdx.x) >> 5;
  int lane  = threadIdx.x & 31;
  int userBase = gwave * 16;
  if (userBase >= kUsers) return;             // wave-uniform (kUsers % 16 == 0)

  int hsel = lane >> 4;
  int li   = lane & 15;
  int user = userBase + li;

  // logits = u_prev[user] . latent[f]
  float a0 = 0.f, a1 = 0.f, a2 = 0.f, a3 = 0.f;
  const float* up = &u_prev[(size_t)user * kCh];
#pragma unroll 4
  for (int c = 0; c < kCh; ++c) {
    float x = up[c];
    a0 += x * s_lat[0 * kCh + c];
    a1 += x * s_lat[1 * kCh + c];
    a2 += x * s_lat[2 * kCh + c];
    a3 += x * s_lat[3 * kCh + c];
  }
  float m  = fmaxf(fmaxf(a0, a1), fmaxf(a2, a3));
  float e0 = expf(a0 - m), e1 = expf(a1 - m), e2 = expf(a2 - m), e3 = expf(a3 - m);
  float inv = 1.f / (e0 + e1 + e2 + e3);
  float p0 = e0 * inv, p1 = e1 * inv, p2 = e2 * inv, p3 = e3 * inv;

  v2f A;
  A.x = hsel ? p2 : p0;
  A.y = hsel ? p3 : p1;

  float upd[4][8];
  float sq[8];
#pragma unroll
  for (int j = 0; j < 8; ++j) sq[j] = 0.f;

#pragma unroll
  for (int t = 0; t < 4; ++t) {
    int n = t * 16 + li;
    v2f B;
    B.x = s_dw[(hsel ? 2 : 0) * kCh + n];
    B.y = s_dw[(hsel ? 3 : 1) * kCh + n];
    v8f C = {};
    C = __builtin_amdgcn_wmma_f32_16x16x4_f32(false, A, false, B,
                                              (short)0, C, false, false);
#pragma unroll
    for (int j = 0; j < 8; ++j) {
      int r = userBase + j + 8 * hsel;
      float agg = u_agg[(size_t)r * kCh + n];
      float u = agg * (1.0f + C[j]);
      upd[t][j] = u;
      sq[j] += u * u;
    }
  }

  // per-row sum of squares: reduce across the 16-lane half (masks < 16)
#pragma unroll
  for (int j = 0; j < 8; ++j) {
    float s = sq[j];
    s += __shfl_xor(s, 1, 32);
    s += __shfl_xor(s, 2, 32);
    s += __shfl_xor(s, 4, 32);
    s += __shfl_xor(s, 8, 32);
    sq[j] = 1.0f / fmaxf(sqrtf(s), 1e-12f);
  }

#pragma unroll
  for (int t = 0; t < 4; ++t) {
    int n = t * 16 + li;
#pragma unroll
    for (int j = 0; j < 8; ++j) {
      int r = userBase + j + 8 * hsel;
      size_t off = (size_t)r * kCh + n;
      float v = upd[t][j] * sq[j];
      u_agg[off] = v;            // becomes u_prev for the next hop
      usr_res[off] += v;         // residual accumulation
    }
  }
}

// ---------------------------------------------------------------------------
// Entity finalize: e = l2norm(e_agg / max(count,1)) in place; ent_res += e.
// One wave per entity row, float2 per lane.
// ---------------------------------------------------------------------------
__global__ void __launch_bounds__(256)
k_ent_finalize(float* __restrict__ e_agg,
               const float* __restrict__ counts,
               float* __restrict__ ent_res) {
  int gwave = (blockIdx.x * blockDim.x + threadIdx.x) >> 5;
  int lane  = threadIdx.x & 31;
  if (gwave >= kEnt) return;
  float inv_dn = 1.0f / fmaxf(counts[gwave], 1.0f);
  size_t base = (size_t)gwave * kCh + lane * 2;
  float2 v = *reinterpret_cast<float2*>(&e_agg[base]);
  v.x *= inv_dn; v.y *= inv_dn;
  float s = v.x * v.x + v.y * v.y;
  s += __shfl_xor(s, 1, 32);
  s += __shfl_xor(s, 2, 32);
  s += __shfl_xor(s, 4, 32);
  s += __shfl_xor(s, 8, 32);
  s += __shfl_xor(s, 16, 32);
  float scale = 1.0f / fmaxf(sqrtf(s), 1e-12f);
  v.x *= scale; v.y *= scale;
  *reinterpret_cast<float2*>(&e_agg[base]) = v;  // e_prev for next hop
  float2* r = reinterpret_cast<float2*>(&ent_res[base]);
  float2 o = *r;
  o.x += v.x; o.y += v.y;
  *r = o;
}

// ---------------------------------------------------------------------------
extern "C" void kernel_launch(void* const* d_in, const int* in_sizes, int n_in,
                              void* d_out, int out_size, void* d_ws, size_t ws_size,
                              hipStream_t stream) {
  const float* user_emb   = (const float*)d_in[0];
  const float* entity_emb = (const float*)d_in[1];
  const float* latent     = (const float*)d_in[2];
  const float* weight     = (const float*)d_in[3];
  const float* att        = (const float*)d_in[4];
  const float* inter_val  = (const float*)d_in[5];
  const int*   edge_index = (const int*)d_in[6];
  const int*   edge_type  = (const int*)d_in[7];
  const int*   inter_row  = (const int*)d_in[8];
  const int*   inter_col  = (const int*)d_in[9];

  float* out     = (float*)d_out;
  float* ent_res = out;
  float* usr_res = out + (size_t)kEnt * kCh;
  float* cor_out = out + (size_t)kEnt * kCh + (size_t)kUsers * kCh;

  const size_t entBytes = (size_t)kEnt * kCh * sizeof(float);    // 64 MB
  const size_t usrBytes = (size_t)kUsers * kCh * sizeof(float);  // 38.4 MB
  char* ws = (char*)d_ws;
  size_t o = 0;
  float* eAgg[2]; float* uAgg[2];
  eAgg[0] = (float*)(ws + o); o += entBytes;
  eAgg[1] = (float*)(ws + o); o += entBytes;
  uAgg[0] = (float*)(ws + o); o += usrBytes;
  uAgg[1] = (float*)(ws + o); o += usrBytes;
  float* counts  = (float*)(ws + o); o += (size_t)kEnt * sizeof(float);
  float* disen_w = (float*)(ws + o); o += (size_t)kF * kCh * sizeof(float);
  (void)ws_size; (void)in_sizes; (void)n_in; (void)out_size;

  // Residuals start at the raw embeddings; cor overwritten by k_small.
  hipMemcpyAsync(ent_res, entity_emb, entBytes, hipMemcpyDeviceToDevice, stream);
  hipMemcpyAsync(usr_res, user_emb, usrBytes, hipMemcpyDeviceToDevice, stream);

  k_small<<<1, 64, 0, stream>>>(att, weight, disen_w, cor_out);

  hipMemsetAsync(counts, 0, (size_t)kEnt * sizeof(float), stream);
  k_count<<<(kEdges + 255) / 256, 256, 0, stream>>>(edge_index, counts);

  const float* e_prev = entity_emb;
  const float* u_prev = user_emb;
  const int scatterThreads = kEdges * 16;
  const int scatterBlocks  = (scatterThreads + 255) / 256;
  const int uwaves  = kUsers / 16;                      // 9375 exact
  const int ublocks = (uwaves * 32 + 255) / 256;
  const int eblocks = (kEnt * 32 + 255) / 256;

  for (int hop = 0; hop < 2; ++hop) {
    float* ea = eAgg[hop];
    float* ua = uAgg[hop];
    hipMemsetAsync(ea, 0, entBytes, stream);
    hipMemsetAsync(ua, 0, usrBytes, stream);

    k_kg_scatter<<<scatterBlocks, 256, 0, stream>>>(e_prev, weight, edge_index,
                                                    edge_type, ea);
    k_inter_scatter<<<scatterBlocks, 256, 0, stream>>>(e_prev, inter_val,
                                                       inter_row, inter_col, ua);
    k_user_finalize<<<ublocks, 256, 0, stream>>>(u_prev, ua, latent, disen_w,
                                                 usr_res);
    k_ent_finalize<<<eblocks, 256, 0, stream>>>(ea, counts, ent_res);

    e_prev = ea;   // normalized in place
    u_prev = ua;   // normalized in place
  }
}